// ModuleSoftsplat_45071386804302
// MI455X (gfx1250) — compile-verified
//
#include <hip/hip_runtime.h>
#include <hip/hip_bf16.h>
#include <stdint.h>

#define Bn 4
#define Cn 64
#define Hn 256
#define Wn 448
#define HWn (Hn * Wn)          /* 114688 */
#define NPIX (Bn * HWn)        /* 458752 */
#define SCAT_TILE 256
#define NORM_TILE 1024

typedef uint32_t u32x4 __attribute__((ext_vector_type(4)));
typedef int      i32x4 __attribute__((ext_vector_type(4)));
typedef int      i32x8 __attribute__((ext_vector_type(8)));

// ---------------------------------------------------------------------------
// No-return device-scope f32 atomic add. The clang atomic statement attribute
// guarantees the backend selects native global_atomic_add_f32 (STOREcnt,
// fire-and-forget) instead of a CAS expansion.
// ---------------------------------------------------------------------------
__device__ __forceinline__ void atomic_add_f32(float* p, float v) {
#if defined(__has_cpp_attribute) && __has_cpp_attribute(clang::atomic)
  [[clang::atomic(no_remote_memory, no_fine_grained_memory, ignore_denormal_mode)]] {
    (void)__hip_atomic_fetch_add(p, v, __ATOMIC_RELAXED, __HIP_MEMORY_SCOPE_AGENT);
  }
#else
  (void)__hip_atomic_fetch_add(p, v, __ATOMIC_RELAXED, __HIP_MEMORY_SCOPE_AGENT);
#endif
}

// ---------------------------------------------------------------------------
// TDM: async 1D tensor load of `n` f32 elements from global -> LDS.
// Descriptor per CDNA5 ISA section 8 (D# groups). Issued once per wave
// (EXEC ignored by TENSOR_LOAD_TO_LDS); caller waits TENSORcnt then barriers.
// ---------------------------------------------------------------------------
__device__ __forceinline__ void tdm_load_f32_1d(uint32_t lds_byte_off,
                                                const float* gptr,
                                                uint32_t n) {
#if __has_builtin(__builtin_amdgcn_tensor_load_to_lds)
  const uint64_t ga = (uint64_t)(uintptr_t)gptr;
  u32x4 g0;
  g0[0] = 1u;                                   // count=1 (valid), user mode
  g0[1] = lds_byte_off;                         // lds_addr (bytes)
  g0[2] = (uint32_t)ga;                         // global_addr[31:0]
  g0[3] = (uint32_t)((ga >> 32) & 0x1FFFFFFu)   // global_addr[56:32]
        | (2u << 30);                           // type = 2 ("image")
  i32x8 g1;
  g1[0] = (int)(2u << 16);                      // data_size = 4 bytes
  g1[1] = (int)((n & 0xFFFFu) << 16);           // tensor_dim0[15:0]
  g1[2] = (int)((n >> 16) & 0xFFFFu)            // tensor_dim0[31:16]
        | (int)(1u << 16);                      // tensor_dim1 = 1
  g1[3] = (int)((n & 0xFFFFu) << 16);           // tile_dim0 = n
  g1[4] = 1;                                    // tile_dim1 = 1, tile_dim2 = 0
  g1[5] = (int)n;                               // tensor_dim0_stride[31:0]
  g1[6] = 0;
  g1[7] = 0;
  const i32x4 z4 = {0, 0, 0, 0};                // groups 2/3 unused (<=2D)
#if __clang_major__ >= 23
  const i32x8 z8 = {0, 0, 0, 0, 0, 0, 0, 0};
  __builtin_amdgcn_tensor_load_to_lds(g0, g1, z4, z4, z8, 0);
#else
  __builtin_amdgcn_tensor_load_to_lds(g0, g1, z4, z4, 0);
#endif
#else
  // Fallback: cooperative direct load by the issuing wave.
  float* l = (float*)((uintptr_t)lds_byte_off); // never taken on gfx1250
  for (uint32_t i = threadIdx.x; i < n; i += 32) l[i] = gptr[i];
#endif
}

__device__ __forceinline__ void tdm_wait_all() {
#if __has_builtin(__builtin_amdgcn_s_wait_tensorcnt)
  __builtin_amdgcn_s_wait_tensorcnt(0);
#endif
}

// ---------------------------------------------------------------------------
// Kernel 1: bilinear forward-splat (scatter-add with no-return f32 atomics).
// One thread per source pixel; 64-channel loop; 4 atomics per channel plus
// 4 atomics into the norm plane. Accumulators are L2-resident (119MB < 192MB).
// ---------------------------------------------------------------------------
__global__ __launch_bounds__(SCAT_TILE) void softsplat_scatter(
    const float* __restrict__ inp,    // [B,64,H,W]
    const float* __restrict__ flow,   // [B,2,H,W]
    const float* __restrict__ metric, // [B,1,H,W]
    float* __restrict__ out,          // [B,64,H,W] accumulators (pre-zeroed)
    float* __restrict__ norm)         // [B,H,W]   accumulators (pre-zeroed)
{
  __shared__ float sfx[SCAT_TILE];
  __shared__ float sfy[SCAT_TILE];
  __shared__ float smt[SCAT_TILE];

  const int blocks_per_batch = HWn / SCAT_TILE;       // 448
  const int b    = blockIdx.x / blocks_per_batch;
  const int pix0 = (blockIdx.x % blocks_per_batch) * SCAT_TILE;
  const int t    = threadIdx.x;

  if (t < 32) {  // wave 0 issues the async tensor loads, then drains TENSORcnt
    tdm_load_f32_1d((uint32_t)(uintptr_t)sfx,
                    flow + (size_t)(b * 2 + 0) * HWn + pix0, SCAT_TILE);
    tdm_load_f32_1d((uint32_t)(uintptr_t)sfy,
                    flow + (size_t)(b * 2 + 1) * HWn + pix0, SCAT_TILE);
    tdm_load_f32_1d((uint32_t)(uintptr_t)smt,
                    metric + (size_t)b * HWn + pix0, SCAT_TILE);
    tdm_wait_all();
  }
  __syncthreads();

  const int pix = pix0 + t;
  const int y = pix / Wn;
  const int x = pix - y * Wn;

  const float fx = (float)x + sfx[t];
  const float fy = (float)y + sfy[t];
  const float m  = expf(smt[t]);

  const float x0f = floorf(fx);
  const float y0f = floorf(fy);
  const int   ix0 = (int)x0f;
  const int   iy0 = (int)y0f;
  const float ax  = fx - x0f;
  const float ay  = fy - y0f;

  float wts[4];
  wts[0] = (1.0f - ax) * (1.0f - ay);
  wts[1] = ax * (1.0f - ay);
  wts[2] = (1.0f - ax) * ay;
  wts[3] = ax * ay;

  int  offs[4];
  bool valid[4];
#pragma unroll
  for (int k = 0; k < 4; ++k) {
    const int cx = ix0 + (k & 1);
    const int cy = iy0 + (k >> 1);
    valid[k] = (cx >= 0) & (cx < Wn) & (cy >= 0) & (cy < Hn);
    offs[k]  = cy * Wn + cx;
  }

  float* nrm = norm + (size_t)b * HWn;
#pragma unroll
  for (int k = 0; k < 4; ++k)
    if (valid[k]) atomic_add_f32(nrm + offs[k], m * wts[k]);

  const float* src = inp + (size_t)b * Cn * HWn + pix;
  float*       dst = out + (size_t)b * Cn * HWn;
  for (int c = 0; c < Cn; ++c) {
    const float v = src[(size_t)c * HWn] * m;  // coalesced per-wave load
#pragma unroll
    for (int k = 0; k < 4; ++k)
      if (valid[k]) atomic_add_f32(dst + (size_t)c * HWn + offs[k], v * wts[k]);
  }
}

// ---------------------------------------------------------------------------
// Kernel 2: out /= (norm==0 ? 1 : norm).  TDM stages a 1024-pixel norm tile
// into LDS once; reused across all 64 channels. One IEEE divide per pixel
// component (reciprocal hoisted), float4 traffic in the channel loop.
// ---------------------------------------------------------------------------
__global__ __launch_bounds__(256) void softsplat_normalize(
    float* __restrict__ out, const float* __restrict__ norm)
{
  __shared__ float snorm[NORM_TILE];

  const int b  = blockIdx.y;
  const int p0 = blockIdx.x * NORM_TILE;

  if (threadIdx.x < 32) {
    tdm_load_f32_1d((uint32_t)(uintptr_t)snorm,
                    norm + (size_t)b * HWn + p0, NORM_TILE);
    tdm_wait_all();
  }
  __syncthreads();

  float4 nv = ((const float4*)snorm)[threadIdx.x];
  const float r0 = 1.0f / ((nv.x == 0.0f) ? 1.0f : nv.x);
  const float r1 = 1.0f / ((nv.y == 0.0f) ? 1.0f : nv.y);
  const float r2 = 1.0f / ((nv.z == 0.0f) ? 1.0f : nv.z);
  const float r3 = 1.0f / ((nv.w == 0.0f) ? 1.0f : nv.w);

  float* base = out + (size_t)b * Cn * HWn + p0;
  for (int c = 0; c < Cn; ++c) {
    float4* row = (float4*)(base + (size_t)c * HWn);
    float4 v = row[threadIdx.x];
    v.x *= r0;
    v.y *= r1;
    v.z *= r2;
    v.w *= r3;
    row[threadIdx.x] = v;
  }
}

// ---------------------------------------------------------------------------
extern "C" void kernel_launch(void* const* d_in, const int* in_sizes, int n_in,
                              void* d_out, int out_size, void* d_ws, size_t ws_size,
                              hipStream_t stream) {
  const float* inp    = (const float*)d_in[0];  // [B,64,H,W]
  const float* flow   = (const float*)d_in[1];  // [B,2,H,W]
  const float* metric = (const float*)d_in[2];  // [B,1,H,W]
  float* out  = (float*)d_out;                  // [B,64,H,W]
  float* norm = (float*)d_ws;                   // [B,H,W] scratch

  (void)in_sizes; (void)n_in; (void)out_size; (void)ws_size;

  hipMemsetAsync(out,  0, (size_t)Bn * Cn * HWn * sizeof(float), stream);
  hipMemsetAsync(norm, 0, (size_t)Bn * HWn * sizeof(float), stream);

  softsplat_scatter<<<NPIX / SCAT_TILE, SCAT_TILE, 0, stream>>>(
      inp, flow, metric, out, norm);

  softsplat_normalize<<<dim3(HWn / NORM_TILE, Bn), 256, 0, stream>>>(out, norm);
}